// LaneDirectionField_87686052316014
// MI455X (gfx1250) — compile-verified
//
#include <hip/hip_runtime.h>

typedef float v2f __attribute__((ext_vector_type(2)));
typedef float v8f __attribute__((ext_vector_type(8)));

#define RES   128
#define GPTS  (RES * RES)   // 16384
#define B_    4
#define L_    64
#define P_    33
#define S_    (L_ * (P_ - 1))  // 2048 segments per batch
#define EPSF  1e-6f
#define EPS2F 1e-12f

__global__ __launch_bounds__(256) void LaneDirectionField_kernel(
    const float* __restrict__ lanes,  // (B, L, P, 2)
    const float* __restrict__ mask,   // (B, L)
    float* __restrict__ out)          // (B, 2, RES, RES)
{
    // Segment tables, LDS-resident for the whole block (one batch per block).
    __shared__ float4 tArr[S_];        // { tx*m, ty*m, m, mx^2+my^2 }
    __shared__ float2 bLowArr[S_];     // { -2*mx, -2*my }
    __shared__ float  stage[8][16][3]; // per-wave finalize staging

    const int tid = threadIdx.x;
    const int b   = blockIdx.x >> 7;   // 128 blocks per batch

    // ---- Build segment tables (each thread: 8 segments) ----
    #pragma unroll
    for (int k = 0; k < S_ / 256; ++k) {
        const int s = tid + (k << 8);
        const int l = s >> 5;          // lane id 0..63
        const int p = s & 31;          // segment id within lane 0..31
        const float* pp = lanes + ((((b * L_ + l) * P_) + p) << 1);
        const float x0 = pp[0], y0 = pp[1], x1 = pp[2], y1 = pp[3];
        const float dx = x1 - x0, dy = y1 - y0;
        const float len = fmaxf(sqrtf(dx * dx + dy * dy), EPSF);
        const float m  = mask[b * L_ + l];
        const float tx = dx / len, ty = dy / len;
        const float mx = 0.5f * (x0 + x1), my = 0.5f * (y0 + y1);
        tArr[s]    = make_float4(tx * m, ty * m, m, mx * mx + my * my);
        bLowArr[s] = make_float2(-2.0f * mx, -2.0f * my);
    }
    __syncthreads();

    const int lane = tid & 31;
    const int half = lane >> 4;   // 0: lanes 0-15 (K0/K1, rows 0-7), 1: lanes 16-31 (K2/K3, rows 8-15)
    const int l15  = lane & 15;
    const int wave = tid >> 5;
    const int tile = ((blockIdx.x & 127) << 3) + wave;   // 0..1023 per batch
    const int g    = (tile << 4) + l15;                  // this lane's grid point (A row)
    const int xi   = g & (RES - 1);
    const int yi   = g >> 7;
    const float gx = -30.0f + (float)xi * (60.0f / 127.0f);
    const float gy = -10.0f + (float)yi * (60.0f / 127.0f);

    // A matrix (16 grid points x 4), constant over the segment loop.
    // d2(g,s) = [gx, gy, gx^2+gy^2, 1] . [-2mx, -2my, 1, mx^2+my^2]
    v2f amat;
    amat.x = half ? (gx * gx + gy * gy) : gx;
    amat.y = half ? 1.0f                : gy;

    float accx[8], accy[8], accd[8];
    #pragma unroll
    for (int r = 0; r < 8; ++r) { accx[r] = 0.f; accy[r] = 0.f; accd[r] = 0.f; }

    // ---- 128 chunks of 16 segments: WMMA distance tile + IDW accumulate ----
    for (int c = 0; c < S_ / 16; ++c) {
        const int s = (c << 4) + l15;          // this lane's segment column
        const float4 tv = tArr[s];
        const float2 bl = bLowArr[s];
        v2f bmat;
        bmat.x = half ? 1.0f : bl.x;
        bmat.y = half ? tv.w : bl.y;
        v8f cz = {0.f, 0.f, 0.f, 0.f, 0.f, 0.f, 0.f, 0.f};
        v8f d2v = __builtin_amdgcn_wmma_f32_16x16x4_f32(
            false, amat, false, bmat, (short)0, cz, false, false);
        #pragma unroll
        for (int r = 0; r < 8; ++r) {
            // dist = max(sqrt(d2), eps); w = 1/dist^2  ==  1/max(d2, eps^2)
            const float w = __builtin_amdgcn_rcpf(fmaxf(d2v[r], EPS2F));
            accx[r] = fmaf(w, tv.x, accx[r]);
            accy[r] = fmaf(w, tv.y, accy[r]);
            accd[r] = fmaf(w, tv.z, accd[r]);
        }
    }

    // ---- Reduce over the 16 segment classes (halves stay independent) ----
    #pragma unroll
    for (int r = 0; r < 8; ++r) {
        #pragma unroll
        for (int msk = 1; msk < 16; msk <<= 1) {
            accx[r] += __shfl_xor(accx[r], msk, 32);
            accy[r] += __shfl_xor(accy[r], msk, 32);
            accd[r] += __shfl_xor(accd[r], msk, 32);
        }
    }
    if (l15 == 0) {
        #pragma unroll
        for (int r = 0; r < 8; ++r) {
            stage[wave][half * 8 + r][0] = accx[r];
            stage[wave][half * 8 + r][1] = accy[r];
            stage[wave][half * 8 + r][2] = accd[r];
        }
    }
    __syncthreads();

    // ---- Finalize: 16 lanes -> 16 grid points of this wave's tile ----
    if (lane < 16) {
        const float nx = stage[wave][l15][0];
        const float ny = stage[wave][l15][1];
        const float dn = fmaxf(stage[wave][l15][2], EPSF);
        float dirx = nx / dn;
        float diry = ny / dn;
        const float nrm = fmaxf(sqrtf(dirx * dirx + diry * diry), EPSF);
        dirx /= nrm;
        diry /= nrm;
        out[(b * 2 + 0) * GPTS + g] = dirx;
        out[(b * 2 + 1) * GPTS + g] = diry;
    }
}

extern "C" void kernel_launch(void* const* d_in, const int* in_sizes, int n_in,
                              void* d_out, int out_size, void* d_ws, size_t ws_size,
                              hipStream_t stream) {
    (void)in_sizes; (void)n_in; (void)d_ws; (void)ws_size; (void)out_size;
    const float* lanes = (const float*)d_in[0];  // (4,64,33,2) f32
    const float* mask  = (const float*)d_in[1];  // (4,64) f32
    float* out = (float*)d_out;                  // (4,2,128,128) f32
    // 4 batches * 128 blocks each; 256 threads = 8 wave32 tiles per block
    LaneDirectionField_kernel<<<512, 256, 0, stream>>>(lanes, mask, out);
}